// EventSampler_11321533792787
// MI455X (gfx1250) — compile-verified
//
#include <hip/hip_runtime.h>
#include <cstdint>
#include <cstddef>

#define BLK_THREADS 128
#define EE 100
#define KK 100
#define MM 10
#define SS 10
#define DTIME_MAX_F 5.0f
#define OVER_RATE 1.5f

// jax.nn.softplus(x) = max(x,0) + log1p(exp(-|x|))  (numerically stable)
__device__ __forceinline__ float softplusf(float x) {
    return fmaxf(x, 0.0f) + log1pf(expf(-fabsf(x)));
}

__device__ __forceinline__ float intens_total_at(float t, float g,
                                                 const float* __restrict__ mu,
                                                 const float* __restrict__ al,
                                                 const float* __restrict__ be) {
    float s = 0.0f;
#pragma unroll
    for (int m = 0; m < MM; ++m) {
        s += softplusf(mu[m] + al[m] * g * expf(-be[m] * t));
    }
    return s;
}

__global__ __launch_bounds__(BLK_THREADS)
void event_sampler_kernel(const int*   __restrict__ type_seqs,  // [B*L]
                          const float* __restrict__ e_unif,     // [B*L, E]
                          const float* __restrict__ u,          // [B*L, K, E]
                          const float* __restrict__ mu,         // [M]
                          const float* __restrict__ al,         // [M]
                          const float* __restrict__ be,         // [M]
                          const float* __restrict__ gam,        // [NTYPES]
                          const int*   __restrict__ nsamp,      // [1]
                          float*       __restrict__ out,        // res [B*L,K] ++ weights [B*L,K]
                          int nbl) {
    __shared__ float s_u[KK * 32];   // first 32 candidate columns of each u row (12.8 KB)
    __shared__ float s_expj[EE];
    __shared__ float s_binv[EE];     // bound / intens_total[e]
    __shared__ float s_tot[SS];
    __shared__ float s_res[KK];

    const int blk  = blockIdx.x;
    const int tid  = threadIdx.x;
    const int lane = tid & 31;
    const int wave = tid >> 5;

    const float* u_bl = u + (size_t)blk * (size_t)(KK * EE);

    // ---- CDNA5 async DMA: stage u[blk, 0:100, 0:32] into LDS (800 x 16B) ----
    // u is stream-once data: non-temporal hint keeps it from parking in L2.
    // Low 32 bits of a generic pointer to __shared__ == LDS byte offset.
    unsigned lds_base = (unsigned)(uintptr_t)(&s_u[0]);
    for (int i = tid; i < KK * 8; i += BLK_THREADS) {
        int k  = i >> 3;         // row
        int c4 = i & 7;          // 16B chunk within first 32 floats
        unsigned long long ga =
            (unsigned long long)(uintptr_t)(u_bl + (size_t)k * EE + (size_t)c4 * 4);
        unsigned la = lds_base + (unsigned)i * 16u;
        asm volatile("global_load_async_to_lds_b128 %0, %1, off th:TH_LOAD_NT"
                     :: "v"(la), "v"(ga) : "memory");
    }

    const float g = gam[type_seqs[blk]];

    // ---- intensity upper bound over S boundary samples (overlaps with DMA) ----
    if (tid < SS) {
        float t = (DTIME_MAX_F * (float)tid) / (float)(SS - 1);
        s_tot[tid] = intens_total_at(t, g, mu, al, be);
    }
    __syncthreads();
    float bound = s_tot[0];
#pragma unroll
    for (int s2 = 1; s2 < SS; ++s2) bound = fmaxf(bound, s_tot[s2]);
    bound *= OVER_RATE;

    // ---- exp_j = cumsum(-log1p(-e)/bound): wave-0 prefix scan over 100 elems ----
    if (tid < 32) {
        float carry = 0.0f;
        for (int c = 0; c < 4; ++c) {
            int idx = c * 32 + lane;
            float v = 0.0f;
            if (idx < EE) v = -log1pf(-e_unif[(size_t)blk * EE + idx]) / bound;
#pragma unroll
            for (int d = 1; d < 32; d <<= 1) {
                float n = __shfl_up(v, d, 32);
                if (lane >= d) v += n;
            }
            if (idx < EE) s_expj[idx] = v + carry;
            carry += __shfl(v, 31, 32);
        }
    }
    __syncthreads();

    // ---- acceptance threshold per candidate: u*bound/total < 1  <=>  u*(bound/total) < 1
    if (tid < EE) {
        s_binv[tid] = bound / intens_total_at(s_expj[tid], g, mu, al, be);
    }

    // drain async DMA, then make all LDS writes visible block-wide
    asm volatile("s_wait_asynccnt 0" ::: "memory");
    __syncthreads();

    // Hoist per-lane candidate constants into registers: the k-loop then only
    // touches LDS for the s_u row read and the one-word result store.
    const float binv_l = s_binv[lane];
    const float expj_l = fminf(s_expj[lane], 1e5f);

    // ---- thinning: each wave handles 25 rows; first-accept via ballot/ctz ----
    for (int j = 0; j < KK / 4; ++j) {
        int k = wave * (KK / 4) + j;
        float uv  = s_u[k * 32 + lane];
        bool pred = uv * binv_l < 1.0f;
        unsigned m = (unsigned)__ballot(pred);
        float res = 0.0f;
        if (m != 0u) {
            res = __shfl(expj_l, __builtin_ctz(m), 32);  // broadcast accepted jump time
        } else {
            // rare path (~1e-5): scan remaining candidates straight from HBM
            for (int c = 1; c < 4; ++c) {
                int e = c * 32 + lane;
                bool p2 = false;
                if (e < EE) {
                    float uv2 = __builtin_nontemporal_load(&u_bl[(size_t)k * EE + e]);
                    p2 = uv2 * s_binv[e] < 1.0f;
                }
                unsigned m2 = (unsigned)__ballot(p2);
                if (m2 != 0u) {
                    res = fminf(s_expj[c * 32 + __builtin_ctz(m2)], 1e5f);
                    break;
                }
            }
        }
        if (lane == 0) s_res[k] = res;
    }
    __syncthreads();

    // ---- coalesced output: res then weights ----
    const float w = 1.0f / (float)nsamp[0];
    if (tid < KK) {
        out[(size_t)blk * KK + tid] = s_res[tid];
        out[(size_t)nbl * KK + (size_t)blk * KK + tid] = w;
    }
}

extern "C" void kernel_launch(void* const* d_in, const int* in_sizes, int n_in,
                              void* d_out, int out_size, void* d_ws, size_t ws_size,
                              hipStream_t stream) {
    // setup_inputs order:
    // 0 time_seqs(f32,unused) 1 time_delta_seqs(f32,unused) 2 type_seqs(i32)
    // 3 e_unif(f32) 4 u(f32) 5 mu 6 alpha 7 beta 8 gamma 9 num_sample(i32 scalar)
    const int*   type_seqs = (const int*)d_in[2];
    const float* e_unif    = (const float*)d_in[3];
    const float* u         = (const float*)d_in[4];
    const float* mu        = (const float*)d_in[5];
    const float* alpha     = (const float*)d_in[6];
    const float* beta      = (const float*)d_in[7];
    const float* gamma_    = (const float*)d_in[8];
    const int*   nsamp     = (const int*)d_in[9];
    float* out = (float*)d_out;

    const int nbl = in_sizes[2];  // B*L = 8192 sites

    event_sampler_kernel<<<dim3(nbl), dim3(BLK_THREADS), 0, stream>>>(
        type_seqs, e_unif, u, mu, alpha, beta, gamma_, nsamp, out, nbl);
}